// feature_selection_layer_6674379178457
// MI455X (gfx1250) — compile-verified
//
#include <hip/hip_runtime.h>
#include <hip/hip_bf16.h>

typedef __attribute__((ext_vector_type(2))) float v2f;
typedef __attribute__((ext_vector_type(8))) float v8f;

#define B_   16
#define N_   20000
#define F_   512
#define K_   200
#define ROWS (B_ * N_)

// ---------------------------------------------------------------------------
// Kernel 1: scores[b,n] = dot(input[b,n,:], kernel[:,0])  via V_WMMA_F32_16X16X4_F32
// One wave = one 16-row tile (M=16). B (4x16) = kernel chunk broadcast across
// all 16 columns -> every column of D holds the score; we read column 0.
// Pure bandwidth kernel: 655 MB streamed, ~28us floor at 23.3 TB/s.
// ---------------------------------------------------------------------------
__global__ __launch_bounds__(256) void scores_wmma_kernel(
    const float* __restrict__ in, const float* __restrict__ kern,
    float* __restrict__ scores)
{
    __shared__ float kbuf[F_];
    const int tid = threadIdx.x;
    kbuf[tid]       = kern[tid];
    kbuf[tid + 256] = kern[tid + 256];
    __syncthreads();

    const int wave = tid >> 5;                 // 8 waves / block
    const int lane = tid & 31;
    const long long tile = (long long)blockIdx.x * 8 + wave;
    const long long row0 = tile * 16;

    const int m  = lane & 15;                  // row within tile (A-matrix M)
    const int kh = (lane >> 4) * 2;            // lane-half K offset {0,2}
    const float* ap = in + (row0 + m) * (long long)F_ + kh;
    const float* bp = kbuf + kh;

    // stream the 2KB row ahead of the MMA loop
    __builtin_prefetch(ap + 128, 0, 1);
    __builtin_prefetch(ap + 256, 0, 1);
    __builtin_prefetch(ap + 384, 0, 1);

    v8f acc = {};
#pragma unroll 8
    for (int k0 = 0; k0 < F_; k0 += 4) {
        v2f a = *(const v2f*)(ap + k0);        // global_load_b64, A layout 16x4
        v2f b = *(const v2f*)(bp + k0);        // ds_load_b64,     B layout 4x16 (bcast)
        acc = __builtin_amdgcn_wmma_f32_16x16x4_f32(
            /*neg_a=*/false, a, /*neg_b=*/false, b,
            /*c_mod=*/(short)0, acc, /*reuse_a=*/false, /*reuse_b=*/false);
    }

    // Column 0 of D: lanes 0 (M=0..7 in VGPR r=M) and 16 (M=8..15 in VGPR r=M-8)
    if ((lane & 15) == 0) {
        const int mb = (lane >> 4) * 8;
#pragma unroll
        for (int r = 0; r < 8; ++r)
            scores[row0 + mb + r] = acc[r];
    }
}

// ---------------------------------------------------------------------------
// Kernel 2: per-batch softmax stats over N=20000 (max, sum of exp)
// ---------------------------------------------------------------------------
__global__ __launch_bounds__(1024) void softmax_stats_kernel(
    const float* __restrict__ scores, float* __restrict__ bmax,
    float* __restrict__ bsum)
{
    __shared__ float red[1024];
    const int b = blockIdx.x, tid = threadIdx.x;
    const float* s = scores + (long long)b * N_;

    float m = -__builtin_inff();
    for (int n = tid; n < N_; n += 1024) m = fmaxf(m, s[n]);
    red[tid] = m; __syncthreads();
    for (int o = 512; o > 0; o >>= 1) {
        if (tid < o) red[tid] = fmaxf(red[tid], red[tid + o]);
        __syncthreads();
    }
    m = red[0]; __syncthreads();

    float sum = 0.f;
    for (int n = tid; n < N_; n += 1024) sum += __expf(s[n] - m);
    red[tid] = sum; __syncthreads();
    for (int o = 512; o > 0; o >>= 1) {
        if (tid < o) red[tid] += red[tid + o];
        __syncthreads();
    }
    if (tid == 0) { bmax[b] = m; bsum[b] = red[0]; }
}

// ---------------------------------------------------------------------------
// Kernel 3: probs = exp(score - max) / sum  (written straight into d_out)
// ---------------------------------------------------------------------------
__global__ __launch_bounds__(256) void probs_kernel(
    const float* __restrict__ scores, const float* __restrict__ bmax,
    const float* __restrict__ bsum, float* __restrict__ probs)
{
    const long long g = (long long)blockIdx.x * 256 + threadIdx.x;
    if (g >= ROWS) return;
    const int b = (int)(g / N_);
    probs[g] = __expf(scores[g] - bmax[b]) / bsum[b];
}

// ---------------------------------------------------------------------------
// Kernel 4: per-batch top-200 via MSB-first radix select on the monotone
// uint transform, then compaction + bitonic sort (descending, like top_k).
// ---------------------------------------------------------------------------
__global__ __launch_bounds__(256) void topk_kernel(
    const float* __restrict__ scores, int* __restrict__ topidx)
{
    __shared__ unsigned hist[256];
    __shared__ unsigned sh_prefix;
    __shared__ int      sh_kk;
    __shared__ int      sidx[256];
    __shared__ unsigned sval[256];
    __shared__ int      cntGT, cntEQ;

    const int b = blockIdx.x, tid = threadIdx.x;
    const float* s = scores + (long long)b * N_;

    if (tid == 0) { sh_prefix = 0u; sh_kk = K_; }

    for (int shift = 24; shift >= 0; shift -= 8) {
        hist[tid] = 0u;
        __syncthreads();
        const unsigned prefix = sh_prefix;
        const unsigned pmask  = (shift == 24) ? 0u : (0xFFFFFFFFu << (shift + 8));
        for (int n = tid; n < N_; n += 256) {
            unsigned fb = __float_as_uint(s[n]);
            unsigned u  = (fb & 0x80000000u) ? ~fb : (fb | 0x80000000u);
            if ((u & pmask) == prefix)
                atomicAdd(&hist[(u >> shift) & 255u], 1u);
        }
        __syncthreads();
        if (tid == 0) {
            int kk = sh_kk;
            unsigned cum = 0;
            for (int d = 255; d >= 0; --d) {
                const unsigned c = hist[d];
                if (cum + c >= (unsigned)kk) {
                    sh_kk = kk - (int)cum;
                    sh_prefix = prefix | ((unsigned)d << shift);
                    break;
                }
                cum += c;
            }
        }
        __syncthreads();
    }

    const unsigned thr = sh_prefix;      // exact threshold key
    const int needEQ   = sh_kk;          // how many ==thr to keep
    const int gtCount  = K_ - needEQ;
    if (tid == 0) { cntGT = 0; cntEQ = 0; }
    sidx[tid] = -1; sval[tid] = 0u;      // sentinels sink in descending sort
    __syncthreads();

    for (int n = tid; n < N_; n += 256) {
        unsigned fb = __float_as_uint(s[n]);
        unsigned u  = (fb & 0x80000000u) ? ~fb : (fb | 0x80000000u);
        if (u > thr) {
            int p = atomicAdd(&cntGT, 1);
            if (p < K_) { sidx[p] = n; sval[p] = u; }
        } else if (u == thr) {
            int p = atomicAdd(&cntEQ, 1);
            if (p < needEQ) { sidx[gtCount + p] = n; sval[gtCount + p] = u; }
        }
    }
    __syncthreads();

    // bitonic sort 256 entries, descending by key
    for (int k = 2; k <= 256; k <<= 1) {
        for (int j = k >> 1; j > 0; j >>= 1) {
            int ixj = tid ^ j;
            if (ixj > tid) {
                bool desc = ((tid & k) == 0);
                unsigned a = sval[tid], c = sval[ixj];
                if (desc ? (a < c) : (a > c)) {
                    sval[tid] = c; sval[ixj] = a;
                    int t = sidx[tid]; sidx[tid] = sidx[ixj]; sidx[ixj] = t;
                }
            }
            __syncthreads();
        }
    }
    if (tid < K_) topidx[b * K_ + tid] = sidx[tid];
}

// ---------------------------------------------------------------------------
// Kernel 5: re-softmax over the 200 selected probs, weighted sum -> [B,F]
// ---------------------------------------------------------------------------
__global__ __launch_bounds__(512) void select_sum_kernel(
    const float* __restrict__ in, const float* __restrict__ probs,
    const int* __restrict__ topidx, float* __restrict__ summed)
{
    __shared__ float w[K_];
    __shared__ int   idx[K_];
    __shared__ float red[256];
    const int b = blockIdx.x, tid = threadIdx.x;

    if (tid < K_) {
        int i = topidx[b * K_ + tid];
        idx[tid] = i;
        w[tid] = probs[(long long)b * N_ + i];
    }
    __syncthreads();

    if (tid < 256) red[tid] = (tid < K_) ? w[tid] : -__builtin_inff();
    __syncthreads();
    for (int o = 128; o > 0; o >>= 1) {
        if (tid < o) red[tid] = fmaxf(red[tid], red[tid + o]);
        __syncthreads();
    }
    const float m = red[0];
    __syncthreads();

    if (tid < 256) red[tid] = (tid < K_) ? __expf(w[tid] - m) : 0.f;
    __syncthreads();
    for (int o = 128; o > 0; o >>= 1) {
        if (tid < o) red[tid] += red[tid + o];
        __syncthreads();
    }
    const float sinv = 1.f / red[0];
    __syncthreads();
    if (tid < K_) w[tid] = __expf(w[tid] - m) * sinv;
    __syncthreads();

    float acc = 0.f;
#pragma unroll 4
    for (int i = 0; i < K_; ++i)
        acc = fmaf(w[i], in[((long long)b * N_ + idx[i]) * F_ + tid], acc);
    summed[b * F_ + tid] = acc;
}

// ---------------------------------------------------------------------------
// Kernel 6: global (axis=None) L2 normalize of summed [B,F]
// ---------------------------------------------------------------------------
__global__ __launch_bounds__(1024) void l2norm_kernel(
    const float* __restrict__ summed, float* __restrict__ out)
{
    __shared__ float red[1024];
    const int tid = threadIdx.x;
    float ssq = 0.f;
    for (int i = tid; i < B_ * F_; i += 1024) {
        float v = summed[i];
        ssq = fmaf(v, v, ssq);
    }
    red[tid] = ssq; __syncthreads();
    for (int o = 512; o > 0; o >>= 1) {
        if (tid < o) red[tid] += red[tid + o];
        __syncthreads();
    }
    const float scale = rsqrtf(fmaxf(red[0], 1e-12f));
    for (int i = tid; i < B_ * F_; i += 1024)
        out[i] = summed[i] * scale;
}

// ---------------------------------------------------------------------------
extern "C" void kernel_launch(void* const* d_in, const int* in_sizes, int n_in,
                              void* d_out, int out_size, void* d_ws, size_t ws_size,
                              hipStream_t stream)
{
    const float* in   = (const float*)d_in[0];   // [B,N,F] f32
    const float* kern = (const float*)d_in[1];   // [F,1]   f32
    // d_in[2] = select_dim (== K_, compiled in)

    float* out    = (float*)d_out;
    int*   topidx = (int*)d_out;                 // [B,K] int32 bits
    float* probs  = out + B_ * K_;               // [B,N,1]
    float* outemb = out + B_ * K_ + B_ * N_;     // [B,F]

    float* scores = (float*)d_ws;                // B*N floats
    float* bmax   = scores + ROWS;               // 16
    float* bsum   = bmax + B_;                   // 16
    float* summed = bsum + B_;                   // B*F floats

    scores_wmma_kernel<<<ROWS / 16 / 8, 256, 0, stream>>>(in, kern, scores);
    softmax_stats_kernel<<<B_, 1024, 0, stream>>>(scores, bmax, bsum);
    probs_kernel<<<(ROWS + 255) / 256, 256, 0, stream>>>(scores, bmax, bsum, probs);
    topk_kernel<<<B_, 256, 0, stream>>>(scores, topidx);
    select_sum_kernel<<<B_, F_, 0, stream>>>(in, probs, topidx, summed);
    l2norm_kernel<<<1, 1024, 0, stream>>>(summed, outemb);
}